// MPIRendererRT_90572270338761
// MI455X (gfx1250) — compile-verified
//
#include <hip/hip_runtime.h>

// MPI bokeh renderer for MI455X (gfx1250).
// Gather + transcendental bound; whole input set (~84MB) is L2-resident
// (192MB L2), HBM traffic is compulsory-only. No GEMM structure -> no WMMA;
// the gfx1250-relevant paths are global_prefetch_b8, batched gathers with
// immediate plane offsets, and nontemporal output stores.

namespace {
constexpr int Wd = 512;
constexpr int Hd = 512;
constexpr int Ld = 8;
constexpr int HWc = Wd * Hd;
constexpr int TPB = 256;
}

__global__ __launch_bounds__(TPB)
void mpi_bokeh_kernel(const float* __restrict__ images,   // (B,L,3,H,W)
                      const float* __restrict__ alphas,   // (B,L,1,H,W)
                      const float* __restrict__ coffs,    // (B,L,1,H,W)
                      const float* __restrict__ Kv,       // (B)
                      const float* __restrict__ dfv,      // (B)
                      const float* __restrict__ etav,     // (B)
                      const float* __restrict__ ellm,     // (B,1,H,W)
                      const float* __restrict__ angm,     // (B,1,H,W)
                      const int*   __restrict__ sppv,     // scalar
                      float* __restrict__ out,            // (B,4,H,W) ++ (B,1,H,W)
                      int B)
{
    const int blocksPerB = HWc / TPB;
    const int b   = blockIdx.x / blocksPerB;           // block-uniform -> s_load constants
    const int pix = (blockIdx.x % blocksPerB) * TPB + threadIdx.x;
    const int px0 = pix & (Wd - 1);
    const int py0 = pix >> 9;

    const float K     = Kv[b];
    const float df    = dfv[b];
    const float eta   = etav[b];
    const float inv2K = 0.5f / K;

    const float ell = ellm[b * HWc + pix];
    const float ang = angm[b * HWc + pix];
    float sa, ca;
    __sincosf(ang, &sa, &ca);

    const float* imgB = images + (size_t)b * Ld * 3 * HWc;
    const float* aB   = alphas + (size_t)b * Ld * HWc;
    const float* cB   = coffs  + (size_t)b * Ld * HWc;

    float col0 = 0.f, col1 = 0.f, col2 = 0.f, accA = 0.f, wsum = 0.f;

    const int   spp  = sppv[0];
    const float step = 2.0f / (float)(spp - 1);

    for (int i = 0; i < spp; ++i) {            // scalar (uniform) sample loop
        const float sx = -1.0f + step * (float)i;
        for (int j = 0; j < spp; ++j) {
            const float sy = -1.0f + step * (float)j;
            const float r2 = sx * sx + sy * sy;
            if (r2 > 1.0f) continue;           // uniform branch: outside disk

            // warped gather coordinate (per-pixel anisotropic kernel)
            const float ux = ca * sx - sa * sy;
            const float uy = (sa * sx + ca * sy) * ell;
            const float rt = __fsqrt_rn(r2) - inv2K;   // inside test: rt <= |c-df|

            float fx = rintf((float)px0 + ux * K);     // round-half-even == jnp.round
            float fy = rintf((float)py0 + uy * K);
            fx = fminf(fmaxf(fx, 0.f), (float)(Wd - 1));
            fy = fminf(fmaxf(fy, 0.f), (float)(Hd - 1));
            const int flat = ((int)fy << 9) | (int)fx;

            // Batch-issue all alpha/coff gathers: 16 independent loads in
            // flight (immediate plane offsets l*1MB fit signed-24b IOFFSET)
            // before the serially-dependent transmittance loop.
            float cl[Ld], al[Ld];
            #pragma unroll
            for (int l = 0; l < Ld; ++l) {
                cl[l] = cB[l * HWc + flat];
                al[l] = aB[l * HWc + flat];
            }
            // Pull each layer's image lines toward the WGP while the math
            // runs (gfx1250 global_prefetch_b8; address is already known).
            #pragma unroll
            for (int l = 0; l < Ld; ++l) {
                __builtin_prefetch(imgB + (size_t)(l * 3) * HWc + flat, 0, 3);
            }

            float T = 1.0f;
            #pragma unroll
            for (int l = 0; l < Ld; ++l) {
                const float ad = fabsf(cl[l] - df);
                if (rt <= ad) {                        // inside circle of confusion
                    const float a   = al[l];
                    const float coc = K * ad;
                    const float w   = T * a * __builtin_amdgcn_rcpf(coc * coc + 1.0f);
                    const float* ip = imgB + (size_t)(l * 3) * HWc + flat;
                    const float i0 = ip[0];
                    const float i1 = ip[HWc];
                    const float i2 = ip[2 * HWc];
                    // img^eta = exp2(eta * log2(img)); img in (0,1)
                    // v_exp_f32 / v_log_f32 via amdgcn builtins (HIP's __exp2f
                    // collides with the glibc host decl on this toolchain).
                    col0 += w * __builtin_amdgcn_exp2f(eta * __builtin_amdgcn_logf(i0));
                    col1 += w * __builtin_amdgcn_exp2f(eta * __builtin_amdgcn_logf(i1));
                    col2 += w * __builtin_amdgcn_exp2f(eta * __builtin_amdgcn_logf(i2));
                    accA += w * a;
                    wsum += w;
                    T    *= 1.0f - a;
                }
            }
        }
    }

    // Write-once outputs: nontemporal so the 10MB result doesn't evict the
    // L2-resident gather working set.
    const size_t o = (size_t)b * 4 * HWc + pix;
    __builtin_nontemporal_store(col0, &out[o]);
    __builtin_nontemporal_store(col1, &out[o + HWc]);
    __builtin_nontemporal_store(col2, &out[o + 2 * HWc]);
    __builtin_nontemporal_store(accA, &out[o + 3 * HWc]);
    __builtin_nontemporal_store(wsum, &out[(size_t)B * 4 * HWc + (size_t)b * HWc + pix]);
}

extern "C" void kernel_launch(void* const* d_in, const int* in_sizes, int n_in,
                              void* d_out, int out_size, void* d_ws, size_t ws_size,
                              hipStream_t stream) {
    (void)n_in; (void)out_size; (void)d_ws; (void)ws_size;
    const float* images = (const float*)d_in[0];
    const float* alphas = (const float*)d_in[1];
    const float* coffs  = (const float*)d_in[2];
    const float* Kv     = (const float*)d_in[3];
    const float* dfv    = (const float*)d_in[4];
    const float* etav   = (const float*)d_in[5];
    const float* ellm   = (const float*)d_in[6];
    const float* angm   = (const float*)d_in[7];
    const int*   sppv   = (const int*)  d_in[8];

    const int B = in_sizes[3];                 // K has shape (B,)
    const int grid = B * (HWc / TPB);          // one thread per pixel, b uniform per block

    mpi_bokeh_kernel<<<grid, TPB, 0, stream>>>(images, alphas, coffs,
                                               Kv, dfv, etav, ellm, angm, sppv,
                                               (float*)d_out, B);
}